// StochasticRGCN_9723805958347
// MI455X (gfx1250) — compile-verified
//
#include <hip/hip_runtime.h>
#include <hip/hip_bf16.h>

// ---------------------------------------------------------------------------
// StochasticRGCN on MI455X (gfx1250, wave32)
//   layer: xw = h @ [W_r0 | W_r1 | loop]  (one fused WMMA f32 GEMM, K=128)
//          agg = scatter_add(xw[src, et]) (f32 atomics, L2-resident)
//          h'  = LN(agg)*g+beta + b + selfloop  (wave32 shuffle reduction)
//   out  : h2 @ Wp + bp                   (WMMA f32 GEMM)
// ---------------------------------------------------------------------------

typedef float v2f __attribute__((ext_vector_type(2)));
typedef float v8f __attribute__((ext_vector_type(8)));

#define NNODES 50000
#define NEDGES 800000
#define DIM    128          // IN = H1 = H2
#define EMB    64
#define NRELS  2
#define LNEPS  1e-5f

#define TILE_M 32
#define TILE_N 64
#define LDK    132          // padded LDS stride (128 + 4) -> conflict-free b64 reads

// ---------------------------------------------------------------------------
// Pack B = [W[r=0] | W[r=1] | loop] into [128][384] row-major (tiny: 192KB)
// ---------------------------------------------------------------------------
__global__ __launch_bounds__(256) void pack_b_kernel(
    const float* __restrict__ W,      // [R][128][128]
    const float* __restrict__ loopw,  // [128][128]
    float* __restrict__ Bp)           // [128][384]
{
    int idx = blockIdx.x * 256 + threadIdx.x;   // 0 .. 49151
    int d = idx / (NRELS * DIM + DIM);
    int c = idx % (NRELS * DIM + DIM);
    float v;
    if (c < NRELS * DIM)
        v = W[((c >> 7) * DIM + d) * DIM + (c & (DIM - 1))];
    else
        v = loopw[d * DIM + (c - NRELS * DIM)];
    Bp[idx] = v;
}

// ---------------------------------------------------------------------------
// Zero fill (float4 granularity)
// ---------------------------------------------------------------------------
__global__ __launch_bounds__(256) void zero_kernel(float* __restrict__ p, long nvec4)
{
    long i = (long)blockIdx.x * 256 + threadIdx.x;
    if (i < nvec4) *(float4*)(p + i * 4) = make_float4(0.f, 0.f, 0.f, 0.f);
}

// ---------------------------------------------------------------------------
// WMMA f32 GEMM:  C[M][NCOLS] = A[M][128] * B[128][NCOLS] (+ bias)
// Block = 256 threads = 8 waves; block tile 32(M) x 64(N); wave tile 16x16.
// Entire K=128 staged in LDS (A row-major padded, B transposed padded).
// ---------------------------------------------------------------------------
__global__ __launch_bounds__(256) void wmma_gemm_f32_kernel(
    const float* __restrict__ A,     // [M][128]
    const float* __restrict__ B,     // [128][NCOLS]
    const float* __restrict__ bias,  // [NCOLS] or nullptr
    float* __restrict__ C,           // [M][NCOLS]
    int M, int NCOLS)
{
    __shared__ float sA[TILE_M * LDK];   // 32 rows x 128 K (padded)
    __shared__ float sBt[TILE_N * LDK];  // 64 cols x 128 K (transposed, padded)

    const int tid  = threadIdx.x;
    const int lane = tid & 31;
    const int wave = tid >> 5;
    const int rowBase = blockIdx.x * TILE_M;
    const int colBase = blockIdx.y * TILE_N;

    // ---- stage A tile: 4096 floats, 4 x float4 per thread, coalesced ----
#pragma unroll
    for (int i = 0; i < 4; ++i) {
        int idx = (i * 256 + tid) * 4;          // float index, 16B aligned
        int r = idx >> 7;                        // row 0..31
        int c = idx & (DIM - 1);
        float4 v = make_float4(0.f, 0.f, 0.f, 0.f);
        int gr = rowBase + r;
        if (gr < M) v = *(const float4*)(A + (long)gr * DIM + c);
        *(float4*)(&sA[r * LDK + c]) = v;
    }
    // ---- stage B tile transposed: 8192 floats, coalesced global reads ----
#pragma unroll
    for (int i = 0; i < 32; ++i) {
        int idx = i * 256 + tid;                 // 0..8191
        int d = idx >> 6;                        // K row 0..127
        int c = idx & (TILE_N - 1);              // col 0..63
        sBt[c * LDK + d] = B[(long)d * NCOLS + (colBase + c)];
    }
    __syncthreads();

    // ---- wave 16x16 tile via 32 x v_wmma_f32_16x16x4_f32 ----
    const int tRow = (wave >> 2) * 16;           // 0 / 16
    const int tCol = (wave & 3) * 16;            // 0 / 16 / 32 / 48
    const int mn   = lane & 15;                  // M index for A frag, N index for B frag
    const int kb   = (lane >> 4) * 2;            // lanes 16-31 carry K+2, K+3

    const float* aptr = &sA [(tRow + mn) * LDK + kb];
    const float* bptr = &sBt[(tCol + mn) * LDK + kb];

    v8f acc = {0.f, 0.f, 0.f, 0.f, 0.f, 0.f, 0.f, 0.f};
#pragma unroll
    for (int k = 0; k < DIM; k += 4) {
        v2f a = *(const v2f*)(aptr + k);   // A[m][k+kb], A[m][k+kb+1]
        v2f b = *(const v2f*)(bptr + k);   // B[k+kb][n], B[k+kb+1][n]
        acc = __builtin_amdgcn_wmma_f32_16x16x4_f32(
            /*neg_a=*/false, a, /*neg_b=*/false, b,
            /*c_mod=*/(short)0, acc, /*reuse_a=*/false, /*reuse_b=*/false);
    }

    // ---- store: VGPR v -> row tRow+v (lanes 0-15) / tRow+v+8 (lanes 16-31) ----
    const int crow = rowBase + tRow + (lane >> 4) * 8;
    const int ccol = colBase + tCol + mn;
    float badd = (bias != nullptr) ? bias[ccol] : 0.f;
#pragma unroll
    for (int v = 0; v < 8; ++v) {
        int gr = crow + v;
        if (gr < M) C[(long)gr * NCOLS + ccol] = acc[v] + badd;
    }
}

// ---------------------------------------------------------------------------
// Edge scatter: one wave per edge; 128 channels -> 4 f32 atomics per lane.
// xw layout: [N][384], relation r message at column r*128.
// ---------------------------------------------------------------------------
__global__ __launch_bounds__(256) void scatter_edges_kernel(
    const int* __restrict__ src, const int* __restrict__ dst,
    const int* __restrict__ et,
    const float* __restrict__ xw,   // [N][384]
    float* __restrict__ agg,        // [N][128]
    int nE)
{
    int e = blockIdx.x * 8 + (threadIdx.x >> 5);
    int lane = threadIdx.x & 31;
    if (e >= nE) return;
    int s = src[e], d = dst[e], r = et[e];
    const float* mp = xw + (long)s * (NRELS * DIM + DIM) + r * DIM;
    float* ap = agg + (long)d * DIM;
#pragma unroll
    for (int i = 0; i < 4; ++i) {
        int c = lane + i * 32;
        atomicAdd(ap + c, mp[c]);       // global_atomic_add_f32, L2-resident
    }
}

// ---------------------------------------------------------------------------
// Fused LayerNorm(affine) + bias + self-loop add. One wave per row (wave32).
// ---------------------------------------------------------------------------
__global__ __launch_bounds__(256) void norm_bias_loop_kernel(
    const float* __restrict__ agg,   // [N][128]
    const float* __restrict__ xw,    // [N][384]; selfloop = cols 256..383
    const float* __restrict__ g, const float* __restrict__ beta,
    const float* __restrict__ b,
    float* __restrict__ out,         // [N][128]
    int n)
{
    int row  = blockIdx.x * 8 + (threadIdx.x >> 5);
    int lane = threadIdx.x & 31;
    if (row >= n) return;

    float4 x = *(const float4*)(agg + (long)row * DIM + lane * 4);
    float s = x.x + x.y + x.z + x.w;
#pragma unroll
    for (int off = 16; off > 0; off >>= 1) s += __shfl_xor(s, off, 32);
    float mu = s * (1.f / DIM);

    float dx = x.x - mu, dy = x.y - mu, dz = x.z - mu, dw = x.w - mu;
    float vs = dx * dx + dy * dy + dz * dz + dw * dw;
#pragma unroll
    for (int off = 16; off > 0; off >>= 1) vs += __shfl_xor(vs, off, 32);
    float rstd = rsqrtf(vs * (1.f / DIM) + LNEPS);

    float4 gg = *(const float4*)(g    + lane * 4);
    float4 be = *(const float4*)(beta + lane * 4);
    float4 bi = *(const float4*)(b    + lane * 4);
    float4 sl = *(const float4*)(xw + (long)row * (NRELS * DIM + DIM) + NRELS * DIM + lane * 4);

    float4 o;
    o.x = dx * rstd * gg.x + be.x + bi.x + sl.x;
    o.y = dy * rstd * gg.y + be.y + bi.y + sl.y;
    o.z = dz * rstd * gg.z + be.z + bi.z + sl.z;
    o.w = dw * rstd * gg.w + be.w + bi.w + sl.w;
    *(float4*)(out + (long)row * DIM + lane * 4) = o;
}

// ---------------------------------------------------------------------------
// Launch
// ---------------------------------------------------------------------------
extern "C" void kernel_launch(void* const* d_in, const int* in_sizes, int n_in,
                              void* d_out, int out_size, void* d_ws, size_t ws_size,
                              hipStream_t stream) {
    (void)in_sizes; (void)n_in; (void)out_size; (void)ws_size;

    const float* h     = (const float*)d_in[0];
    const int*   src1  = (const int*)  d_in[1];
    const int*   dst1  = (const int*)  d_in[2];
    const int*   et1   = (const int*)  d_in[3];
    const int*   src2  = (const int*)  d_in[4];
    const int*   dst2  = (const int*)  d_in[5];
    const int*   et2   = (const int*)  d_in[6];
    const float* W1    = (const float*)d_in[7];
    const float* loop1 = (const float*)d_in[8];
    const float* b1    = (const float*)d_in[9];
    const float* g1    = (const float*)d_in[10];
    const float* beta1 = (const float*)d_in[11];
    const float* W2    = (const float*)d_in[12];
    const float* loop2 = (const float*)d_in[13];
    const float* b2    = (const float*)d_in[14];
    const float* g2    = (const float*)d_in[15];
    const float* beta2 = (const float*)d_in[16];
    const float* Wp    = (const float*)d_in[17];
    const float* bp    = (const float*)d_in[18];
    float* out = (float*)d_out;

    // workspace carve-up (floats)
    const long NB  = (long)DIM * (NRELS * DIM + DIM);   // 49152   packed B
    const long NXW = (long)NNODES * (NRELS * DIM + DIM);// 19.2M   xw + selfloop
    const long NH  = (long)NNODES * DIM;                // 6.4M
    float* ws  = (float*)d_ws;
    float* Bp  = ws;
    float* xw  = Bp + NB;
    float* agg = xw + NXW;
    float* h1  = agg + NH;
    float* h2  = h1 + NH;

    const int gM    = (NNODES + TILE_M - 1) / TILE_M;   // 1563
    const int gRows = (NNODES + 7) / 8;                 // 6250 (norm)
    const int gZero = (int)(NH / 4 / 256);              // 6250
    const int gEdge = (NEDGES + 7) / 8;                 // 100000
    const int gPack = (int)(NB / 256);                  // 192

    // ---------------- layer 1 ----------------
    pack_b_kernel<<<gPack, 256, 0, stream>>>(W1, loop1, Bp);
    zero_kernel<<<gZero, 256, 0, stream>>>(agg, NH / 4);
    wmma_gemm_f32_kernel<<<dim3(gM, (NRELS * DIM + DIM) / TILE_N), 256, 0, stream>>>(
        h, Bp, nullptr, xw, NNODES, NRELS * DIM + DIM);
    scatter_edges_kernel<<<gEdge, 256, 0, stream>>>(src1, dst1, et1, xw, agg, NEDGES);
    norm_bias_loop_kernel<<<gRows, 256, 0, stream>>>(agg, xw, g1, beta1, b1, h1, NNODES);

    // ---------------- layer 2 ----------------
    pack_b_kernel<<<gPack, 256, 0, stream>>>(W2, loop2, Bp);
    zero_kernel<<<gZero, 256, 0, stream>>>(agg, NH / 4);
    wmma_gemm_f32_kernel<<<dim3(gM, (NRELS * DIM + DIM) / TILE_N), 256, 0, stream>>>(
        h1, Bp, nullptr, xw, NNODES, NRELS * DIM + DIM);
    scatter_edges_kernel<<<gEdge, 256, 0, stream>>>(src2, dst2, et2, xw, agg, NEDGES);
    norm_bias_loop_kernel<<<gRows, 256, 0, stream>>>(agg, xw, g2, beta2, b2, h2, NNODES);

    // ---------------- projection ----------------
    wmma_gemm_f32_kernel<<<dim3(gM, 1), 256, 0, stream>>>(
        h2, Wp, bp, out, NNODES, EMB);
}